// ScalarAttention_59579786330696
// MI455X (gfx1250) — compile-verified
//
#include <hip/hip_runtime.h>
#include <hip/hip_bf16.h>
#include <stdint.h>

typedef __attribute__((ext_vector_type(2))) float v2f;
typedef __attribute__((ext_vector_type(8))) float v8f;

#define Tn    512
#define TD    256
#define TPR   12
#define HID   32
#define INNER 256
#define BN_EPS 1e-5f
#define ATT_SCALE 0.0625f   // 256^-0.5

// Workspace layout (floats)
#define WS_W1F   0              // 12*32 = 384
#define WS_O1    384            // 32
#define WS_W2F   416            // 32*256 = 8192
#define WS_O2    8608           // 256
#define WS_XB    8864           // 2*512*256 = 262144
#define WS_QKV   (8864+262144)  // 2*512*768 = 786432

// ---------------------------------------------------------------- prep: fold BN into weights
__global__ __launch_bounds__(256) void prep_kernel(
    const float* __restrict__ w1, const float* __restrict__ b1,
    const float* __restrict__ g1, const float* __restrict__ be1,
    const float* __restrict__ m1, const float* __restrict__ v1,
    const float* __restrict__ w2, const float* __restrict__ b2,
    const float* __restrict__ g2, const float* __restrict__ be2,
    const float* __restrict__ m2, const float* __restrict__ v2,
    float* __restrict__ ws)
{
    int tid = threadIdx.x;
    float* w1f = ws + WS_W1F;
    float* o1  = ws + WS_O1;
    float* w2f = ws + WS_W2F;
    float* o2  = ws + WS_O2;
    if (tid < HID) {
        float s = g1[tid] * rsqrtf(v1[tid] + BN_EPS);
        o1[tid] = (b1[tid] - m1[tid]) * s + be1[tid];
    }
    {
        float s = g2[tid] * rsqrtf(v2[tid] + BN_EPS);
        o2[tid] = (b2[tid] - m2[tid]) * s + be2[tid];
    }
    for (int i = tid; i < TPR * HID; i += 256) {
        int j = i & (HID - 1);
        w1f[i] = w1[i] * (g1[j] * rsqrtf(v1[j] + BN_EPS));
    }
    for (int i = tid; i < HID * TD; i += 256) {
        int c = i & (TD - 1);
        w2f[i] = w2[i] * (g2[c] * rsqrtf(v2[c] + BN_EPS));
    }
}

// ---------------------------------------------------------------- fused TPR MLP + max + add-x
// One workgroup per (b,s). 8 waves; wave w owns channels [32w, 32w+32).
// r is streamed HBM->LDS with async-to-LDS double buffering, hidden behind WMMA.
#define H1S 36   // padded LDS row stride for h1: bank-conflict-free AND 16B aligned
__global__ __launch_bounds__(256) void tpr_bias_kernel(
    const float* __restrict__ r, const float* __restrict__ x,
    const float* __restrict__ ws, float* __restrict__ xb)
{
    const float* w1f = ws + WS_W1F;
    const float* o1  = ws + WS_O1;
    const float* w2f = ws + WS_W2F;
    const float* o2  = ws + WS_O2;

    __shared__ float sw1[TPR * HID + HID];    // folded w1 + o1 (1664 B)
    __shared__ float sh1[256 * H1S];          // h1 tile, one phase (36864 B)
    __shared__ float rstage[2][256 * TPR];    // async r double buffer (24576 B)

    const int bs   = blockIdx.x;              // b*512 + s
    const int tid  = threadIdx.x;
    const int lane = tid & 31;
    const int wave = tid >> 5;
    const int half = lane >> 4;
    const int l16  = lane & 15;

    const float* rbase = r + (size_t)bs * Tn * TPR;

    // ---- kick off async HBM->LDS stream of r for phase 0 (48 B per thread)
    {
        unsigned lds0 = (unsigned)(uintptr_t)&rstage[0][tid * TPR];
        unsigned long long g0 = (unsigned long long)(rbase + (size_t)tid * TPR);
        asm volatile(
            "global_load_async_to_lds_b128 %0, %1, off\n\t"
            "global_load_async_to_lds_b128 %0, %1, off offset:16\n\t"
            "global_load_async_to_lds_b128 %0, %1, off offset:32"
            :: "v"(lds0), "v"(g0) : "memory");
    }

    for (int i = tid; i < TPR * HID + HID; i += 256)
        sw1[i] = (i < TPR * HID) ? w1f[i] : o1[i - TPR * HID];

    // Preload B fragments (w2f) for this wave's two 16-wide N tiles.
    v2f Bf[2][8];
    float o2v[2];
#pragma unroll
    for (int nt = 0; nt < 2; ++nt) {
        const int n0 = wave * 32 + nt * 16;
#pragma unroll
        for (int kc = 0; kc < 8; ++kc) {
            const int k = kc * 4 + 2 * half;
            Bf[nt][kc].x = w2f[(k + 0) * TD + n0 + l16];
            Bf[nt][kc].y = w2f[(k + 1) * TD + n0 + l16];
        }
        o2v[nt] = o2[n0 + l16];
    }

    v8f vmax[2];
#pragma unroll
    for (int nt = 0; nt < 2; ++nt)
#pragma unroll
        for (int v = 0; v < 8; ++v) vmax[nt][v] = 0.0f;

    asm volatile("s_wait_asynccnt 0x0" ::: "memory");
    __syncthreads();   // rstage[0] + sw1 visible to all waves

    for (int phase = 0; phase < 2; ++phase) {
        // ---- layer 1 (K=12, VALU): one t-row per thread, LDS -> LDS
        {
            const float4* rp4 = reinterpret_cast<const float4*>(&rstage[phase][tid * TPR]);
            float rv[TPR];
            float4 a = rp4[0], b = rp4[1], c = rp4[2];
            rv[0]=a.x; rv[1]=a.y; rv[2]=a.z; rv[3]=a.w;
            rv[4]=b.x; rv[5]=b.y; rv[6]=b.z; rv[7]=b.w;
            rv[8]=c.x; rv[9]=c.y; rv[10]=c.z; rv[11]=c.w;
            float h[HID];
#pragma unroll
            for (int j = 0; j < HID; ++j) {
                float acc = sw1[TPR * HID + j];
#pragma unroll
                for (int d = 0; d < TPR; ++d) acc = fmaf(rv[d], sw1[d * HID + j], acc);
                h[j] = fmaxf(acc, 0.0f);
            }
            float4* dst = reinterpret_cast<float4*>(&sh1[tid * H1S]);
#pragma unroll
            for (int j4 = 0; j4 < 8; ++j4)
                dst[j4] = make_float4(h[4*j4], h[4*j4+1], h[4*j4+2], h[4*j4+3]);
        }
        __syncthreads();   // sh1 ready; rstage[phase] no longer needed

        // ---- prefetch next phase's r while the matrix pipe runs
        if (phase == 0) {
            unsigned lds1 = (unsigned)(uintptr_t)&rstage[1][tid * TPR];
            unsigned long long g1a =
                (unsigned long long)(rbase + (size_t)(256 + tid) * TPR);
            asm volatile(
                "global_load_async_to_lds_b128 %0, %1, off\n\t"
                "global_load_async_to_lds_b128 %0, %1, off offset:16\n\t"
                "global_load_async_to_lds_b128 %0, %1, off offset:32"
                :: "v"(lds1), "v"(g1a) : "memory");
        }

        // ---- layer 2: WMMA f32 16x16x4, 16 t-tiles this phase
        for (int tt = 0; tt < 16; ++tt) {
            v2f Af[8];
#pragma unroll
            for (int kc = 0; kc < 8; ++kc) {
                const float* p = &sh1[(tt * 16 + l16) * H1S + kc * 4 + 2 * half];
                Af[kc].x = p[0];
                Af[kc].y = p[1];
            }
#pragma unroll
            for (int nt = 0; nt < 2; ++nt) {
                v8f c = {};
#pragma unroll
                for (int kc = 0; kc < 8; ++kc)
                    c = __builtin_amdgcn_wmma_f32_16x16x4_f32(
                        false, Af[kc], false, Bf[nt][kc], (short)0, c, false, false);
#pragma unroll
                for (int v = 0; v < 8; ++v)
                    vmax[nt][v] = fmaxf(vmax[nt][v], c[v] + o2v[nt]);
            }
        }
        if (phase == 0)
            asm volatile("s_wait_asynccnt 0x0" ::: "memory");
        __syncthreads();
    }

    // ---- reduce running max over the 8 row-VGPRs, then across lane halves
#pragma unroll
    for (int nt = 0; nt < 2; ++nt) {
        float m = vmax[nt][0];
#pragma unroll
        for (int v = 1; v < 8; ++v) m = fmaxf(m, vmax[nt][v]);
        m = fmaxf(m, __shfl_xor(m, 16, 32));
        if (half == 0) {
            const int c = wave * 32 + nt * 16 + l16;
            xb[(size_t)bs * TD + c] = x[(size_t)bs * TD + c] + m;
        }
    }
}

// ---------------------------------------------------------------- qkv = xb @ w_qkv  (WMMA f32)
// One workgroup per 16-row M tile; each wave computes 6 of the 48 N tiles.
#define SA_S 260
__global__ __launch_bounds__(256) void qkv_kernel(
    const float* __restrict__ xb, const float* __restrict__ wqkv, float* __restrict__ qkv)
{
    __shared__ float sA[16 * SA_S];
    const int tid  = threadIdx.x;
    const int lane = tid & 31;
    const int wave = tid >> 5;
    const int half = lane >> 4;
    const int l16  = lane & 15;
    const int m0   = blockIdx.x * 16;

    for (int i = tid; i < 16 * TD; i += 256) {
        int row = i >> 8, k = i & 255;
        sA[row * SA_S + k] = xb[(size_t)(m0 + row) * TD + k];
    }
    __syncthreads();

    for (int j = 0; j < 6; ++j) {
        const int n0 = (wave + 8 * j) * 16;
        v8f c = {};
        for (int kc = 0; kc < 64; ++kc) {
            const int k = kc * 4 + 2 * half;
            v2f a, b;
            a.x = sA[l16 * SA_S + k];
            a.y = sA[l16 * SA_S + k + 1];
            b.x = wqkv[(size_t)(k + 0) * (3 * INNER) + n0 + l16];
            b.y = wqkv[(size_t)(k + 1) * (3 * INNER) + n0 + l16];
            c = __builtin_amdgcn_wmma_f32_16x16x4_f32(
                false, a, false, b, (short)0, c, false, false);
        }
#pragma unroll
        for (int v = 0; v < 8; ++v) {
            const int row = v + 8 * half;
            qkv[(size_t)(m0 + row) * (3 * INNER) + n0 + l16] = c[v];
        }
    }
}

// ---------------------------------------------------------------- attention (VALU), 1 WG per (b,t)
__global__ __launch_bounds__(256) void attn_kernel(
    const float* __restrict__ qkv, float* __restrict__ out)
{
    __shared__ float sq[TD];
    __shared__ float sp[Tn];
    __shared__ float red[256];

    const int bt = blockIdx.x;
    const int b  = bt >> 9;
    const int tid = threadIdx.x;

    const float* q     = qkv + (size_t)bt * (3 * INNER);
    const float* kbase = qkv + (size_t)b * Tn * (3 * INNER) + INNER;
    const float* vbase = kbase + INNER;

    sq[tid] = q[tid];
    __syncthreads();

    float d0[2];
#pragma unroll
    for (int i = 0; i < 2; ++i) {
        const int s = tid + i * 256;
        const float* kp = kbase + (size_t)s * (3 * INNER);
        float acc = 0.0f;
        for (int c = 0; c < TD; ++c) acc = fmaf(sq[c], kp[c], acc);
        d0[i] = acc * ATT_SCALE;
    }

    // block-wide max
    red[tid] = fmaxf(d0[0], d0[1]);
    __syncthreads();
    for (int off = 128; off > 0; off >>= 1) {
        if (tid < off) red[tid] = fmaxf(red[tid], red[tid + off]);
        __syncthreads();
    }
    const float mx = red[0];
    __syncthreads();

    const float e0 = __expf(d0[0] - mx);
    const float e1 = __expf(d0[1] - mx);
    sp[tid] = e0; sp[tid + 256] = e1;
    red[tid] = e0 + e1;
    __syncthreads();
    for (int off = 128; off > 0; off >>= 1) {
        if (tid < off) red[tid] += red[tid + off];
        __syncthreads();
    }
    const float inv = 1.0f / red[0];
    __syncthreads();

    float acc = 0.0f;
    for (int s = 0; s < Tn; ++s)
        acc = fmaf(sp[s], vbase[(size_t)s * (3 * INNER) + tid], acc);
    out[(size_t)bt * INNER + tid] = acc * inv;
}

// ----------------------------------------------------------------
extern "C" void kernel_launch(void* const* d_in, const int* in_sizes, int n_in,
                              void* d_out, int out_size, void* d_ws, size_t ws_size,
                              hipStream_t stream) {
    (void)in_sizes; (void)n_in; (void)out_size; (void)ws_size;
    const float* x     = (const float*)d_in[0];
    const float* r     = (const float*)d_in[1];
    const float* w1    = (const float*)d_in[2];
    const float* b1    = (const float*)d_in[3];
    const float* g1    = (const float*)d_in[4];
    const float* be1   = (const float*)d_in[5];
    const float* m1    = (const float*)d_in[6];
    const float* v1    = (const float*)d_in[7];
    const float* w2    = (const float*)d_in[8];
    const float* b2    = (const float*)d_in[9];
    const float* g2    = (const float*)d_in[10];
    const float* be2   = (const float*)d_in[11];
    const float* m2    = (const float*)d_in[12];
    const float* v2    = (const float*)d_in[13];
    const float* w_qkv = (const float*)d_in[14];
    float* ws  = (float*)d_ws;
    float* out = (float*)d_out;

    float* xbp  = ws + WS_XB;
    float* qkvp = ws + WS_QKV;

    prep_kernel<<<1, 256, 0, stream>>>(w1, b1, g1, be1, m1, v1,
                                       w2, b2, g2, be2, m2, v2, ws);
    tpr_bias_kernel<<<2 * Tn, 256, 0, stream>>>(r, x, ws, xbp);
    qkv_kernel<<<(2 * Tn) / 16, 256, 0, stream>>>(xbp, w_qkv, qkvp);
    attn_kernel<<<2 * Tn, 256, 0, stream>>>(qkvp, out);
}